// LATTE_58050777973562
// MI455X (gfx1250) — compile-verified
//
#include <hip/hip_runtime.h>

// ---------------------------------------------------------------------------
// Problem constants (match reference)
// ---------------------------------------------------------------------------
#define NN 100000
#define EE 1600000
#define DD 128
#define HH 4
#define CC 32
#define NEG_SLOPE 0.2f

typedef float v2f __attribute__((ext_vector_type(2)));
typedef float v8f __attribute__((ext_vector_type(8)));

__device__ __forceinline__ float lrelu(float v) {
    return v >= 0.0f ? v : NEG_SLOPE * v;
}

// monotonic uint mapping for f32 atomic max
__device__ __forceinline__ unsigned fmap(float f) {
    unsigned u = __float_as_uint(f);
    return (u & 0x80000000u) ? ~u : (u | 0x80000000u);
}
__device__ __forceinline__ float funmap(unsigned u) {
    return (u & 0x80000000u) ? __uint_as_float(u & 0x7FFFFFFFu)
                             : __uint_as_float(~u);
}

// ---------------------------------------------------------------------------
// Kernel 1: dual GEMM  l = x@Wl + bl,  r = x@Wr + br  via V_WMMA_F32_16X16X4_F32
// One wave (32 lanes) per 16x16 output tile; computes BOTH l and r tiles so the
// A operand (rows of x) is loaded once. K=128 -> 32 WMMA steps per accumulator.
// ---------------------------------------------------------------------------
__global__ __launch_bounds__(32) void gemm_lr_wmma(
    const float* __restrict__ x,
    const float* __restrict__ Wl, const float* __restrict__ bl,
    const float* __restrict__ Wr, const float* __restrict__ br,
    float* __restrict__ l, float* __restrict__ r)
{
    const int lane = threadIdx.x;          // 0..31, wave32
    const int m0   = blockIdx.x * 16;      // node-row tile
    const int n0   = blockIdx.y * 16;      // output-column tile
    const int half = lane >> 4;            // 0: lanes 0-15, 1: lanes 16-31
    const int l15  = lane & 15;

    // A (16x4 f32): lanes 0-15 hold M=lane, K={kb,kb+1}; lanes 16-31 M=lane-16, K={kb+2,kb+3}
    const int arow  = m0 + l15;
    const int akoff = half * 2;
    // B (4x16 f32): mirrored striping, N = l15
    const int bcol  = n0 + l15;

    const float* xrow = x + (size_t)arow * DD + akoff;

    v8f accL = {};
    v8f accR = {};

    #pragma unroll 4
    for (int kb = 0; kb < DD; kb += 4) {
        v2f a = *(const v2f*)(xrow + kb);            // contiguous pair, 8B aligned

        const int krow = kb + akoff;
        v2f bL, bR;
        bL.x = Wl[(size_t)(krow    ) * DD + bcol];
        bL.y = Wl[(size_t)(krow + 1) * DD + bcol];
        bR.x = Wr[(size_t)(krow    ) * DD + bcol];
        bR.y = Wr[(size_t)(krow + 1) * DD + bcol];

        accL = __builtin_amdgcn_wmma_f32_16x16x4_f32(
            /*neg_a=*/false, a, /*neg_b=*/false, bL,
            /*c_mod=*/(short)0, accL, /*reuse_a=*/false, /*reuse_b=*/false);
        accR = __builtin_amdgcn_wmma_f32_16x16x4_f32(
            false, a, false, bR, (short)0, accR, false, false);
    }

    // C/D layout: VGPR i -> (M = i + half*8, N = l15)
    const float biasL = bl[bcol];
    const float biasR = br[bcol];
    #pragma unroll
    for (int i = 0; i < 8; ++i) {
        const size_t row = (size_t)(m0 + i + half * 8);
        l[row * DD + bcol] = accL[i] + biasL;
        r[row * DD + bcol] = accR[i] + biasR;
    }
}

// ---------------------------------------------------------------------------
// Kernel 2: per-edge attention logits + segment max (atomic on mapped uints).
// One wave per edge: 8 lanes per head, 4 channels per lane (float4 loads).
// ---------------------------------------------------------------------------
__global__ __launch_bounds__(256) void edge_alpha(
    const int* __restrict__ ei,          // [2,E]
    const float* __restrict__ l, const float* __restrict__ r,
    const float* __restrict__ attn,      // [H, 2C]
    float* __restrict__ outAlpha,        // [E, H] (raw logits this pass)
    unsigned* __restrict__ amax)         // [N, H] mapped
{
    const int w    = threadIdx.x >> 5;
    const int lane = threadIdx.x & 31;
    const size_t e = (size_t)blockIdx.x * 8 + w;
    if (e >= EE) return;

    const int src = ei[e];
    const int dst = ei[(size_t)EE + e];
    const int h   = lane >> 3;
    const int sub = lane & 7;
    const int co  = h * CC + sub * 4;

    const float4 xi = *(const float4*)(r + (size_t)dst * DD + co);
    const float4 xj = *(const float4*)(l + (size_t)src * DD + co);
    const float4 ai = *(const float4*)(attn + h * (2 * CC) + sub * 4);
    const float4 aj = *(const float4*)(attn + h * (2 * CC) + CC + sub * 4);

    float p = ai.x * lrelu(xi.x) + ai.y * lrelu(xi.y)
            + ai.z * lrelu(xi.z) + ai.w * lrelu(xi.w)
            + aj.x * lrelu(xj.x) + aj.y * lrelu(xj.y)
            + aj.z * lrelu(xj.z) + aj.w * lrelu(xj.w);

    // width-8 butterfly reduction (all 8 lanes end with the head sum)
    p += __shfl_xor(p, 4, 32);
    p += __shfl_xor(p, 2, 32);
    p += __shfl_xor(p, 1, 32);

    if (sub == 0) {
        outAlpha[e * HH + h] = p;
        atomicMax(&amax[(size_t)dst * HH + h], fmap(p));
    }
}

// ---------------------------------------------------------------------------
// Kernel 3: exp(alpha - max) + segment sum. One thread per (edge, head).
// ---------------------------------------------------------------------------
__global__ __launch_bounds__(256) void edge_exp(
    const int* __restrict__ ei,
    float* __restrict__ outAlpha,        // in: logits, out: exp terms
    const unsigned* __restrict__ amax,
    float* __restrict__ asum)            // [N, H]
{
    const size_t t = (size_t)blockIdx.x * 256 + threadIdx.x;
    if (t >= (size_t)EE * HH) return;
    const size_t e = t >> 2;             // HH == 4
    const int    h = (int)(t & 3);
    const int  dst = ei[(size_t)EE + e];

    const unsigned u = amax[(size_t)dst * HH + h];
    const float m = (u == 0u) ? 0.0f : funmap(u);   // empty segment -> 0 (isfinite fixup)
    const float ex = __expf(outAlpha[t] - m);
    outAlpha[t] = ex;
    atomicAdd(&asum[(size_t)dst * HH + h], ex);
}

// ---------------------------------------------------------------------------
// Kernel 4: normalize alpha + scatter-add messages x_j * alpha into h_rel.
// One wave per edge; 4 f32 atomics per lane into L2-resident h_rel.
// ---------------------------------------------------------------------------
__global__ __launch_bounds__(256) void edge_scatter(
    const int* __restrict__ ei,
    const float* __restrict__ l,
    const float* __restrict__ asum,
    float* __restrict__ outAlpha,        // in: exp terms, out: normalized alpha
    float* __restrict__ h_rel)           // [N, H, C] accumulated
{
    const int w    = threadIdx.x >> 5;
    const int lane = threadIdx.x & 31;
    const size_t e = (size_t)blockIdx.x * 8 + w;
    if (e >= EE) return;

    const int src = ei[e];
    const int dst = ei[(size_t)EE + e];
    const int h   = lane >> 3;
    const int sub = lane & 7;
    const int co  = h * CC + sub * 4;

    const float aexp = outAlpha[e * HH + h];
    const float s    = asum[(size_t)dst * HH + h];
    const float wgt  = aexp / (s + 1e-16f);
    if (sub == 0) outAlpha[e * HH + h] = wgt;       // final alpha output

    const float4 xj = *(const float4*)(l + (size_t)src * DD + co);
    float* base = h_rel + (size_t)dst * DD + co;
    atomicAdd(base + 0, xj.x * wgt);
    atomicAdd(base + 1, xj.y * wgt);
    atomicAdd(base + 2, xj.z * wgt);
    atomicAdd(base + 3, xj.w * wgt);
}

// ---------------------------------------------------------------------------
// Kernel 5: per-node beta weights over {h_rel, l(self)} + relu output.
// One wave per node; 8 lanes per head.
// ---------------------------------------------------------------------------
__global__ __launch_bounds__(256) void node_out(
    const float* __restrict__ l, const float* __restrict__ r,
    const float* __restrict__ h_rel,
    const float* __restrict__ ral,       // rel_attn_l [H,C]
    const float* __restrict__ rar,       // rel_attn_r [H,C]
    float* __restrict__ out_h)           // [N, D]
{
    const int w    = threadIdx.x >> 5;
    const int lane = threadIdx.x & 31;
    const size_t n = (size_t)blockIdx.x * 8 + w;
    if (n >= NN) return;

    const int h   = lane >> 3;
    const int sub = lane & 7;
    const int co  = h * CC + sub * 4;

    const float4 rv = *(const float4*)(r     + n * DD + co);
    const float4 lv = *(const float4*)(l     + n * DD + co);
    const float4 hv = *(const float4*)(h_rel + n * DD + co);
    const float4 wl = *(const float4*)(ral + co);
    const float4 wr = *(const float4*)(rar + co);

    float pl  = rv.x * wl.x + rv.y * wl.y + rv.z * wl.z + rv.w * wl.w;
    float pr0 = hv.x * wr.x + hv.y * wr.y + hv.z * wr.z + hv.w * wr.w;
    float pr1 = lv.x * wr.x + lv.y * wr.y + lv.z * wr.z + lv.w * wr.w;

    pl  += __shfl_xor(pl,  4, 32); pl  += __shfl_xor(pl,  2, 32); pl  += __shfl_xor(pl,  1, 32);
    pr0 += __shfl_xor(pr0, 4, 32); pr0 += __shfl_xor(pr0, 2, 32); pr0 += __shfl_xor(pr0, 1, 32);
    pr1 += __shfl_xor(pr1, 4, 32); pr1 += __shfl_xor(pr1, 2, 32); pr1 += __shfl_xor(pr1, 1, 32);

    const float b0 = lrelu(pl + pr0);       // relation 0: metapath agg (h_rel)
    const float b1 = lrelu(pl + pr1);       // relation 1: self (l)
    const float m  = fmaxf(b0, b1);
    const float e0 = __expf(b0 - m);
    const float e1 = __expf(b1 - m);
    const float inv = 1.0f / (e0 + e1);
    const float beta0 = e0 * inv;
    const float beta1 = e1 * inv;

    float4 o;
    o.x = fmaxf(hv.x * beta0 + lv.x * beta1, 0.0f);
    o.y = fmaxf(hv.y * beta0 + lv.y * beta1, 0.0f);
    o.z = fmaxf(hv.z * beta0 + lv.z * beta1, 0.0f);
    o.w = fmaxf(hv.w * beta0 + lv.w * beta1, 0.0f);
    *(float4*)(out_h + n * DD + co) = o;
}

// ---------------------------------------------------------------------------
// Launch
// ---------------------------------------------------------------------------
extern "C" void kernel_launch(void* const* d_in, const int* in_sizes, int n_in,
                              void* d_out, int out_size, void* d_ws, size_t ws_size,
                              hipStream_t stream)
{
    const float* x    = (const float*)d_in[0];
    const int*   ei   = (const int*)  d_in[1];
    const float* Wl   = (const float*)d_in[2];
    const float* bl   = (const float*)d_in[3];
    const float* Wr   = (const float*)d_in[4];
    const float* br   = (const float*)d_in[5];
    const float* attn = (const float*)d_in[6];
    const float* ral  = (const float*)d_in[7];
    const float* rar  = (const float*)d_in[8];

    // d_out = [h : N*D floats][alpha : E*H floats]
    float* out_h     = (float*)d_out;
    float* out_alpha = out_h + (size_t)NN * DD;

    // workspace layout
    float*    l     = (float*)d_ws;
    float*    r     = l + (size_t)NN * DD;
    float*    h_rel = r + (size_t)NN * DD;
    unsigned* amax  = (unsigned*)(h_rel + (size_t)NN * DD);
    float*    asum  = (float*)(amax + (size_t)NN * HH);

    // per-call init of accumulation buffers (deterministic across replays)
    hipMemsetAsync(h_rel, 0, (size_t)NN * DD * sizeof(float), stream);
    hipMemsetAsync(amax,  0, (size_t)NN * HH * sizeof(unsigned), stream);
    hipMemsetAsync(asum,  0, (size_t)NN * HH * sizeof(float), stream);

    dim3 ggemm(NN / 16, DD / 16);                       // 6250 x 8 tiles
    gemm_lr_wmma<<<ggemm, 32, 0, stream>>>(x, Wl, bl, Wr, br, l, r);

    edge_alpha<<<EE / 8, 256, 0, stream>>>(ei, l, r, attn, out_alpha, amax);
    edge_exp<<<((size_t)EE * HH + 255) / 256, 256, 0, stream>>>(ei, out_alpha, amax, asum);
    edge_scatter<<<EE / 8, 256, 0, stream>>>(ei, l, asum, out_alpha, h_rel);
    node_out<<<(NN + 7) / 8, 256, 0, stream>>>(l, r, h_rel, ral, rar, out_h);
}